// GumbelTopK_1348619731468
// MI455X (gfx1250) — compile-verified
//
#include <hip/hip_runtime.h>
#include <stdint.h>
#include <stddef.h>
#include <math.h>

#define GN    8192          // row length N
#define KSEL  64u           // top-k
#define TPB   512           // 16 wave32 waves
#define NWAVE (TPB / 32)

typedef float    v4f   __attribute__((ext_vector_type(4)));
typedef unsigned v4u   __attribute__((ext_vector_type(4)));
typedef int      i32x4 __attribute__((ext_vector_type(4)));

#if defined(__HIP_DEVICE_COMPILE__) && __has_builtin(__builtin_amdgcn_global_load_async_to_lds_b128)
#define HAVE_ASYNC_BUILTIN 1
#else
#define HAVE_ASYNC_BUILTIN 0
#endif

#if defined(__HIP_DEVICE_COMPILE__) && __has_builtin(__builtin_amdgcn_s_wait_asynccnt)
#define HAVE_WAIT_BUILTIN 1
#else
#define HAVE_WAIT_BUILTIN 0
#endif

// CDNA5 async global->LDS copy (16B per lane), tracked with ASYNCcnt.
__device__ __forceinline__ void async_copy_b128(const void* gp, void* lp) {
#if HAVE_ASYNC_BUILTIN
  __builtin_amdgcn_global_load_async_to_lds_b128(
      (__attribute__((address_space(1))) i32x4*)gp,
      (__attribute__((address_space(3))) i32x4*)lp,
      0, 0);
#else
  unsigned lds_off = (unsigned)(size_t)(__attribute__((address_space(3))) void*)lp;
  unsigned long long ga = (unsigned long long)(size_t)gp;
  asm volatile("global_load_async_to_lds_b128 %0, %1, off"
               :: "v"(lds_off), "v"(ga)
               : "memory");
#endif
}

__device__ __forceinline__ void wait_async_zero() {
#if HAVE_WAIT_BUILTIN
  __builtin_amdgcn_s_wait_asynccnt(0);
#else
  asm volatile("s_wait_asynccnt 0" ::: "memory");
#endif
}

// Order-preserving float<->uint32 key (ascending key == ascending float).
__device__ __forceinline__ unsigned key_of(float f) {
  unsigned b = __float_as_uint(f);
  return (b & 0x80000000u) ? ~b : (b | 0x80000000u);
}
__device__ __forceinline__ float float_of(unsigned k) {
  unsigned b = (k & 0x80000000u) ? (k ^ 0x80000000u) : ~k;
  return __uint_as_float(b);
}

// Online softmax accumulator update.
__device__ __forceinline__ void online_update(float g, float& m, float& s) {
  if (g > m) { s *= __expf(m - g); m = g; }
  s += __expf(g - m);
}

__global__ __launch_bounds__(TPB) void gumbel_topk_hard_kernel(
    const float* __restrict__ logits,
    const float* __restrict__ noise,
    float* __restrict__ out)
{
  __shared__ alignas(16) unsigned kbuf[GN];   // 32 KB: noise staging, then sort keys
  __shared__ unsigned histw[NWAVE][256];      // 16 KB: wave-private radix histograms
  __shared__ unsigned histsum[256];
  __shared__ float red_m[NWAVE], red_s[NWAVE];
  __shared__ float sh_M, sh_invS;
  __shared__ unsigned sh_prefix, sh_r, sh_takeall;

  const int tid  = threadIdx.x;
  const int lane = tid & 31;
  const int wav  = tid >> 5;
  const size_t base = (size_t)blockIdx.x * GN;
  const float* gl = logits + base;
  const float* gn = noise + base;
  float* go = out + base;

  // ---- stage the gumbel-noise row into LDS with async b128 copies ----
  #pragma unroll
  for (int i = 0; i < 4; ++i) {
    int v = tid + i * TPB;                 // vec4 index, interleaved -> coalesced
    async_copy_b128(gn + 4 * v, &kbuf[4 * v]);
  }
  wait_async_zero();
  __syncthreads();

  // ---- pass 1: g = logits + noise; sortable keys -> LDS; online (max, sum-exp) ----
  float m = -INFINITY, s = 0.0f;
  #pragma unroll
  for (int i = 0; i < 4; ++i) {
    int v = tid + i * TPB;
    v4f a = ((const v4f*)gl)[v];
    v4f b = ((v4f*)kbuf)[v];               // staged noise
    v4f g = a + b;                         // tau == 1
    v4u k;
    k.x = key_of(g.x); k.y = key_of(g.y); k.z = key_of(g.z); k.w = key_of(g.w);
    ((v4u*)kbuf)[v] = k;                   // same slots this thread just read
    online_update(g.x, m, s);
    online_update(g.y, m, s);
    online_update(g.z, m, s);
    online_update(g.w, m, s);
  }

  // wave32 reduction of (m, s)
  #pragma unroll
  for (int off = 16; off > 0; off >>= 1) {
    float om = __shfl_xor(m, off, 32);
    float os = __shfl_xor(s, off, 32);
    float mm = fmaxf(m, om);
    s = s * __expf(m - mm) + os * __expf(om - mm);
    m = mm;
  }
  if (lane == 0) { red_m[wav] = m; red_s[wav] = s; }
  __syncthreads();
  if (tid == 0) {
    float M = red_m[0], S = red_s[0];
    for (int w = 1; w < NWAVE; ++w) {
      float om = red_m[w], os = red_s[w];
      float mm = fmaxf(M, om);
      S = S * __expf(M - mm) + os * __expf(om - mm);
      M = mm;
    }
    sh_M = M;
    sh_invS = 1.0f / S;
    sh_prefix = 0u;
    sh_r = KSEL;
    sh_takeall = 0u;
  }
  __syncthreads();

  // ---- radix-256 select of the 64th-largest key (4 rounds, all in LDS) ----
  for (int round = 0; round < 4; ++round) {
    const unsigned prefix = sh_prefix;
    const unsigned rcur   = sh_r;
    const int shift = 24 - 8 * round;
    const unsigned maskhi = (round == 0) ? 0u : (0xFFFFFFFFu << (8 * (4 - round)));
    for (int t = tid; t < NWAVE * 256; t += TPB) (&histw[0][0])[t] = 0u;
    __syncthreads();
    #pragma unroll
    for (int i = 0; i < 16; ++i) {
      unsigned k = kbuf[tid + i * TPB];
      if ((k & maskhi) == prefix)
        atomicAdd(&histw[wav][(k >> shift) & 0xFFu], 1u);
    }
    __syncthreads();
    if (tid < 256) {
      unsigned c = 0;
      for (int w = 0; w < NWAVE; ++w) c += histw[w][tid];
      histsum[tid] = c;
    }
    __syncthreads();
    if (tid == 0) {
      unsigned cum = 0;
      int d = 255;
      for (; d > 0; --d) {                 // descending rank scan
        unsigned c = histsum[d];
        if (cum + c >= rcur) break;
        cum += c;
      }
      sh_prefix = prefix | ((unsigned)d << shift);
      sh_r = rcur - cum;
      if (round == 3) sh_takeall = (histsum[d] == (rcur - cum)) ? 1u : 0u;
    }
    __syncthreads();
  }

  const unsigned T       = sh_prefix;   // key of the 64th-largest g
  const unsigned need    = sh_r;        // how many T-valued keys to take
  const unsigned takeall = sh_takeall;  // common case: take every key == T
  const float M    = sh_M;
  const float invS = sh_invS;

  // ---- output: (1 - y) + y at top-64, exactly +0.0 elsewhere ----
  #pragma unroll
  for (int i = 0; i < 4; ++i) {
    int v = tid + i * TPB;
    v4u k = ((v4u*)kbuf)[v];
    v4f o;
    #pragma unroll
    for (int c = 0; c < 4; ++c) {
      unsigned kc = (c == 0) ? k.x : (c == 1) ? k.y : (c == 2) ? k.z : k.w;
      float oc = 0.0f;
      bool sel = false;
      if (kc > T) sel = true;
      else if (kc == T) {
        if (takeall) sel = true;
        else {
          // rare tie path: jax top_k prefers lower index among equal values
          int j = 4 * v + c;
          unsigned rank = 0;
          for (int q = 0; q < j; ++q) rank += (kbuf[q] == T) ? 1u : 0u;
          sel = (rank < need);
        }
      }
      if (sel) {
        float y = __expf(float_of(kc) - M) * invS;   // y = softmax(g)[j]
        oc = (1.0f - y) + y;                          // straight-through forward value
      }
      if (c == 0) o.x = oc; else if (c == 1) o.y = oc; else if (c == 2) o.z = oc; else o.w = oc;
    }
    __builtin_nontemporal_store(o, (v4f*)go + v);     // streaming store, keep L2 for inputs
  }
}

extern "C" void kernel_launch(void* const* d_in, const int* in_sizes, int n_in,
                              void* d_out, int out_size, void* d_ws, size_t ws_size,
                              hipStream_t stream) {
  (void)n_in; (void)out_size; (void)d_ws; (void)ws_size;
  const float* logits = (const float*)d_in[0];
  const float* noise  = (const float*)d_in[1];
  float* out = (float*)d_out;
  const int B = in_sizes[0] / GN;        // 2048 rows
  gumbel_topk_hard_kernel<<<dim3((unsigned)B), dim3(TPB), 0, stream>>>(logits, noise, out);
}